// DualPathAttention_15109694947381
// MI455X (gfx1250) — compile-verified
//
#include <hip/hip_runtime.h>
#include <math.h>

// ---------------------------------------------------------------------------
// DualPathAttention for MI455X (gfx1250), fp32 end-to-end via V_WMMA_F32_16X16X4_F32.
// B=2, T=2048, D=1024, H=16, dh=64.  ~150 GFLOP, working set ~160MB (< 192MB L2).
// Tile staging uses GLOBAL_LOAD_ASYNC_TO_LDS_B128 (ASYNCcnt), double-buffered
// so the async DMA overlaps the WMMA work in both the GEMM and attention loops.
// ---------------------------------------------------------------------------

typedef float v2f __attribute__((ext_vector_type(2)));
typedef float v8f __attribute__((ext_vector_type(8)));

#define BB 2
#define TT 2048
#define DDIM 1024
#define HH 16
#define DH 64
#define BT (BB * TT) /* 4096 */

static constexpr size_t SZ_HEAD = (size_t)BB * HH * TT * DH; // 4,194,304 floats

// D = A(16x4) * B(4x16) + C, fp32 WMMA.
// A frag: lane (grp g, l16): a.x = A[l16][k0+2g], a.y = A[l16][k0+2g+1]
// B frag: lane (grp g, l16): b.x = B[k0+2g][l16], b.y = B[k0+2g+1][l16]
// C/D frag: reg r, lane (g,l16): C[r + 8g][l16]
static __device__ __forceinline__ v8f wmma4(v2f a, v2f b, v8f c) {
  return __builtin_amdgcn_wmma_f32_16x16x4_f32(false, a, false, b, (short)0, c,
                                               false, false);
}

static __device__ __forceinline__ v8f v8zero() {
  v8f z = {0.f, 0.f, 0.f, 0.f, 0.f, 0.f, 0.f, 0.f};
  return z;
}

// Low 32 bits of a shared-aperture flat address = wave-relative LDS byte offset.
static __device__ __forceinline__ unsigned lds_off(const void* p) {
  return (unsigned)(unsigned long long)p;
}

// Async memory -> LDS copy of 16 bytes (tracked by ASYNCcnt).
static __device__ __forceinline__ void async_b128(unsigned lds, const float* g) {
  asm volatile("global_load_async_to_lds_b128 %0, %1, off"
               :
               : "v"(lds), "v"(g)
               : "memory");
}
static __device__ __forceinline__ void wait_async0() {
  asm volatile("s_wait_asynccnt 0x0" ::: "memory");
}

// ---------------------------------------------------------------------------
// Generic fp32 WMMA GEMM: C = A(MxK) @ B(KxN) + bias.
// Block = 256 threads (8 waves). Block tile = 128 rows x 64 cols.
// Each wave: 16 rows x 64 cols (4 accumulators). B tile async-staged in LDS,
// double-buffered so the next tile's DMA overlaps the current tile's WMMAs.
// mode 0: row-major C (MxN)
// mode 1: qkv scatter: col -> (which,h,d), row -> (b,t);  C base = q|k|v (B,H,T,dh)
// mode 2: head scatter: col -> (h,d), row -> (b,t);       C = (B,H,T,dh)
// ---------------------------------------------------------------------------
__global__ __launch_bounds__(256) void gemm_f32_wmma(
    const float* __restrict__ A, const float* __restrict__ Bw,
    const float* __restrict__ bias, float* __restrict__ C, int M, int K, int N,
    int mode) {
  __shared__ float Bs[2][16 * 72]; // stride 72: 16B rows for b128

  const int wave = threadIdx.x >> 5;
  const int lane = threadIdx.x & 31;
  const int grp = lane >> 4;
  const int l16 = lane & 15;
  const int row0 = blockIdx.y * 128 + wave * 16;
  const int col0 = blockIdx.x * 64;
  const bool full = (col0 + 64) <= N; // partial only for the N=16 gate GEMM

  v8f acc[4];
  for (int t = 0; t < 4; ++t) acc[t] = v8zero();

  const float* Arow = A + (size_t)(row0 + l16) * K;
  const unsigned bs_base = lds_off(&Bs[0][0]);
  const unsigned bs_stride = 16 * 72 * 4;
  const int li = threadIdx.x;
  const int lr = li >> 4;        // 0..15  (B-tile row)
  const int lc = (li & 15) << 2; // 0..60  (B-tile col, x4 floats)
  const unsigned doff = (unsigned)((lr * 72 + lc) * 4);

  auto stageB = [&](int kb, int bufi) {
    if (full) {
      async_b128(bs_base + (unsigned)bufi * bs_stride + doff,
                 Bw + (size_t)(kb + lr) * N + col0 + lc);
    } else {
      for (int i = li; i < 16 * 64; i += 256) {
        int r = i >> 6, c = i & 63;
        int col = col0 + c;
        Bs[bufi][r * 72 + c] = (col < N) ? Bw[(size_t)(kb + r) * N + col] : 0.f;
      }
    }
  };

  const int nkb = K >> 4;
  stageB(0, 0);
  wait_async0();
  __syncthreads();

  for (int ib = 0; ib < nkb; ++ib) {
    const int cur = ib & 1;
    if (ib + 1 < nkb) stageB((ib + 1) << 4, cur ^ 1); // overlap DMA w/ WMMA
    const float* bs = &Bs[cur][0];
    const int kb = ib << 4;
#pragma unroll
    for (int kk = 0; kk < 16; kk += 4) {
      v2f a = *(const v2f*)(Arow + kb + kk + 2 * grp);
#pragma unroll
      for (int t = 0; t < 4; ++t) {
        v2f b;
        b.x = bs[(kk + 2 * grp) * 72 + t * 16 + l16];
        b.y = bs[(kk + 2 * grp + 1) * 72 + t * 16 + l16];
        acc[t] = wmma4(a, b, acc[t]);
      }
    }
    wait_async0();
    __syncthreads();
  }

  for (int t = 0; t < 4; ++t) {
    int colt = col0 + t * 16;
    if (colt >= N) break; // N is always a multiple of 16 here
    int col = colt + l16;
    float bi = bias ? bias[col] : 0.f;
#pragma unroll
    for (int r = 0; r < 8; ++r) {
      int row = row0 + r + 8 * grp;
      float val = acc[t][r] + bi;
      if (mode == 0) {
        C[(size_t)row * N + col] = val;
      } else if (mode == 1) {
        int which = col >> 10;
        int h = (col >> 6) & 15;
        int d = col & 63;
        int b_ = row >> 11;
        int tk = row & 2047;
        C[(size_t)which * SZ_HEAD +
          (((size_t)(b_ * HH + h)) * TT + tk) * DH + d] = val;
      } else {
        int h = col >> 6;
        int d = col & 63;
        int b_ = row >> 11;
        int tk = row & 2047;
        C[(((size_t)(b_ * HH + h)) * TT + tk) * DH + d] = val;
      }
    }
  }
}

// ---------------------------------------------------------------------------
// Plücker lines + J6 + gate. One block of 32 threads per token.
// jw8 / read8: (B,H,T,8) with last two lanes zero (pad K=6 -> 8 for WMMA).
// Jw = L @ J6 = [L5, -L4, L3, L2, -L1, L0].
// ---------------------------------------------------------------------------
__global__ __launch_bounds__(32) void prep_lines_gate(
    const float* __restrict__ p1w, const float* __restrict__ w2w,
    const float* __restrict__ r1w, const float* __restrict__ r2w,
    const float* __restrict__ gatepre, float* __restrict__ jw8,
    float* __restrict__ read8, float* __restrict__ gate) {
  int token = blockIdx.x; // 0..BT-1
  int b = token >> 11, t = token & 2047;
  int h = threadIdx.x;
  if (h < 16) {
    float p1[4], p2[4], L[6];
    // ---- write lines: exterior(x_prev @ w1_write, x @ w2_write) ----
#pragma unroll
    for (int i = 0; i < 4; ++i)
      p1[i] = (t > 0) ? p1w[(size_t)(token - 1) * 64 + h * 4 + i] : 0.f;
#pragma unroll
    for (int i = 0; i < 4; ++i) p2[i] = w2w[(size_t)token * 64 + h * 4 + i];
    L[0] = p1[0] * p2[1] - p1[1] * p2[0];
    L[1] = p1[0] * p2[2] - p1[2] * p2[0];
    L[2] = p1[0] * p2[3] - p1[3] * p2[0];
    L[3] = p1[1] * p2[2] - p1[2] * p2[1];
    L[4] = p1[1] * p2[3] - p1[3] * p2[1];
    L[5] = p1[2] * p2[3] - p1[3] * p2[2];
    float s = 0.f;
#pragma unroll
    for (int i = 0; i < 6; ++i) s += L[i] * L[i];
    float d = fmaxf(sqrtf(s), 1e-12f);
    float* dj = jw8 + (((size_t)(b * HH + h)) * TT + t) * 8;
    dj[0] = L[5] / d;
    dj[1] = -L[4] / d;
    dj[2] = L[3] / d;
    dj[3] = L[2] / d;
    dj[4] = -L[1] / d;
    dj[5] = L[0] / d;
    dj[6] = 0.f;
    dj[7] = 0.f;
    // ---- read lines: exterior(x @ w1_read, x @ w2_read) ----
#pragma unroll
    for (int i = 0; i < 4; ++i) p1[i] = r1w[(size_t)token * 64 + h * 4 + i];
#pragma unroll
    for (int i = 0; i < 4; ++i) p2[i] = r2w[(size_t)token * 64 + h * 4 + i];
    L[0] = p1[0] * p2[1] - p1[1] * p2[0];
    L[1] = p1[0] * p2[2] - p1[2] * p2[0];
    L[2] = p1[0] * p2[3] - p1[3] * p2[0];
    L[3] = p1[1] * p2[2] - p1[2] * p2[1];
    L[4] = p1[1] * p2[3] - p1[3] * p2[1];
    L[5] = p1[2] * p2[3] - p1[3] * p2[2];
    s = 0.f;
#pragma unroll
    for (int i = 0; i < 6; ++i) s += L[i] * L[i];
    d = fmaxf(sqrtf(s), 1e-12f);
    float* dr = read8 + (((size_t)(b * HH + h)) * TT + t) * 8;
#pragma unroll
    for (int i = 0; i < 6; ++i) dr[i] = L[i] / d;
    dr[6] = 0.f;
    dr[7] = 0.f;
  }
  if (h == 0) {
    float acc = 0.f;
#pragma unroll
    for (int i = 0; i < 16; ++i) {
      float z = gatepre[(size_t)token * 16 + i];
      acc += 1.f / (1.f + __expf(-z));
    }
    gate[token] = acc * (1.f / 16.f);
  }
}

// ---------------------------------------------------------------------------
// Flash-style causal attention, both paths, fp32 WMMA.
// Block = 8 waves = 8 consecutive 16-query tiles of one (b,h). The block runs
// a shared key-tile loop: K_j/V_j/geoV_j/Jw_j are async-staged into LDS
// (double-buffered; DMA for tile j+1 overlaps compute on tile j), then each
// wave with qtile >= j does WMMA logits + online softmax + WMMA PV. P converts
// C-layout -> A-layout via a private LDS slab.
// ---------------------------------------------------------------------------
#define TS 68 /* LDS stride for 16x64 tiles: 16B rows, conflict-free frags */

static __device__ __forceinline__ void softmax_pv(v8f& s, float* m, float* l,
                                                  v8f* o, float* slab,
                                                  const float* __restrict__ Vs,
                                                  int grp, int l16) {
  float alpha[8];
#pragma unroll
  for (int r = 0; r < 8; ++r) {
    float rowmax = s[r];
#pragma unroll
    for (int off = 8; off >= 1; off >>= 1)
      rowmax = fmaxf(rowmax, __shfl_xor(rowmax, off, 32)); // 16-lane half
    float mn = fmaxf(m[r], rowmax);
    alpha[r] = __expf(m[r] - mn);
    m[r] = mn;
    float p = __expf(s[r] - mn);
    s[r] = p;
    float rs = p;
#pragma unroll
    for (int off = 8; off >= 1; off >>= 1) rs += __shfl_xor(rs, off, 32);
    l[r] = l[r] * alpha[r] + rs;
  }
#pragma unroll
  for (int nt = 0; nt < 4; ++nt)
#pragma unroll
    for (int r = 0; r < 8; ++r) o[nt][r] *= alpha[r];

  // C-layout P -> LDS (row r+8g, col l16); stride 18 keeps b64 alignment.
#pragma unroll
  for (int r = 0; r < 8; ++r) slab[(r + 8 * grp) * 18 + l16] = s[r];

  // O += P(16x16) @ V(16x64): 4 k-chunks x 4 n-tiles, V from LDS tile.
#pragma unroll
  for (int kc = 0; kc < 4; ++kc) {
    v2f pa = *(const v2f*)(slab + l16 * 18 + kc * 4 + 2 * grp);
    int kk0 = kc * 4 + 2 * grp;
#pragma unroll
    for (int nt = 0; nt < 4; ++nt) {
      v2f vb;
      vb.x = Vs[kk0 * TS + nt * 16 + l16];
      vb.y = Vs[(kk0 + 1) * TS + nt * 16 + l16];
      o[nt] = wmma4(pa, vb, o[nt]);
    }
  }
}

__global__ __launch_bounds__(256) void attn_dual(
    const float* __restrict__ q, const float* __restrict__ k,
    const float* __restrict__ v, const float* __restrict__ geov,
    const float* __restrict__ read8, const float* __restrict__ jw8,
    const float* __restrict__ gate, const float* __restrict__ inc_scale,
    float* __restrict__ comb) {
  __shared__ float Ks[2][16 * TS];
  __shared__ float Vs[2][16 * TS];
  __shared__ float Gs[2][16 * TS];
  __shared__ float Js[2][16 * 8];
  __shared__ float slab[8 * 16 * 18]; // per-wave P bounce slabs

  const int bh = blockIdx.x; // 0..B*H-1
  const int h = bh & 15;
  const int b = bh >> 4;
  const int wave = threadIdx.x >> 5;
  const int lane = threadIdx.x & 31;
  const int grp = lane >> 4;
  const int l16 = lane & 15;
  const int qtile = blockIdx.y * 8 + wave;
  const int qbase = qtile * 16;
  const int jmax = blockIdx.y * 8 + 7; // block-shared key-tile bound

  const float* qp = q + (size_t)bh * TT * DH;
  const float* kp = k + (size_t)bh * TT * DH;
  const float* vp = v + (size_t)bh * TT * DH;
  const float* gvp = geov + (size_t)bh * TT * DH;
  const float* rp = read8 + (size_t)bh * TT * 8;
  const float* jp = jw8 + (size_t)bh * TT * 8;
  const float inc = inc_scale[h];
  float* myslab = slab + wave * 16 * 18;

  const unsigned ldsK = lds_off(&Ks[0][0]);
  const unsigned ldsV = lds_off(&Vs[0][0]);
  const unsigned ldsG = lds_off(&Gs[0][0]);
  const unsigned ldsJ = lds_off(&Js[0][0]);
  const unsigned tstride = 16 * TS * 4;
  const unsigned jstride = 16 * 8 * 4;
  const int li = threadIdx.x;
  const int lr = li >> 4;        // 0..15 tile row
  const int lc = (li & 15) << 2; // 0..60 tile col (x4 floats)
  const unsigned dtile = (unsigned)((lr * TS + lc) * 4);

  auto stageT = [&](int jb, int bufi) {
    const unsigned bo = (unsigned)bufi * tstride;
    async_b128(ldsK + bo + dtile, kp + (size_t)(jb + lr) * DH + lc);
    async_b128(ldsV + bo + dtile, vp + (size_t)(jb + lr) * DH + lc);
    async_b128(ldsG + bo + dtile, gvp + (size_t)(jb + lr) * DH + lc);
    if (li < 32) {
      int rj = li >> 1, cj = (li & 1) << 2;
      async_b128(ldsJ + (unsigned)bufi * jstride + (unsigned)((rj * 8 + cj) * 4),
                 jp + (size_t)(jb + rj) * 8 + cj);
    }
  };

  // Q fragments (A-layout), 16 k-chunks over dh=64
  v2f qa[16];
#pragma unroll
  for (int kc = 0; kc < 16; ++kc)
    qa[kc] = *(const v2f*)(qp + (size_t)(qbase + l16) * DH + kc * 4 + 2 * grp);
  // read-line fragments, 2 k-chunks over padded K=8
  v2f ra[2];
#pragma unroll
  for (int kc = 0; kc < 2; ++kc)
    ra[kc] = *(const v2f*)(rp + (size_t)(qbase + l16) * 8 + kc * 4 + 2 * grp);

  v8f ostd[4], ogeo[4];
  for (int t = 0; t < 4; ++t) { ostd[t] = v8zero(); ogeo[t] = v8zero(); }
  float ms[8], ls[8], mg[8], lg[8];
#pragma unroll
  for (int r = 0; r < 8; ++r) {
    ms[r] = mg[r] = -3.0e38f;
    ls[r] = lg[r] = 0.f;
  }

  stageT(0, 0);
  wait_async0();
  __syncthreads();

  for (int j = 0; j <= jmax; ++j) {
    const int cur = j & 1;
    if (j < jmax) stageT((j + 1) * 16, cur ^ 1); // overlap DMA w/ compute

    if (j <= qtile) {
      const float* ks = &Ks[cur][0];
      const float* js = &Js[cur][0];
      // ---- Sstd = Q @ K^T * scale ----
      v8f s = v8zero();
#pragma unroll
      for (int kc = 0; kc < 16; ++kc) {
        v2f kb = *(const v2f*)(ks + l16 * TS + kc * 4 + 2 * grp);
        s = wmma4(qa[kc], kb, s);
      }
#pragma unroll
      for (int r = 0; r < 8; ++r) s[r] *= 0.125f; // dh^-0.5

      // ---- Sgeo = read @ Jw^T * inc ----
      v8f sg = v8zero();
#pragma unroll
      for (int kc = 0; kc < 2; ++kc) {
        v2f jbv = *(const v2f*)(js + l16 * 8 + kc * 4 + 2 * grp);
        sg = wmma4(ra[kc], jbv, sg);
      }
#pragma unroll
      for (int r = 0; r < 8; ++r) sg[r] *= inc;

      if (j == qtile) { // causal mask on the diagonal tile (k > q masked)
#pragma unroll
        for (int r = 0; r < 8; ++r) {
          int qr = r + 8 * grp;
          s[r] = (l16 > qr) ? -3.0e38f : s[r];
          sg[r] = (l16 > qr) ? -3.0e38f : sg[r];
        }
      }

      softmax_pv(s, ms, ls, ostd, myslab, &Vs[cur][0], grp, l16);
      softmax_pv(sg, mg, lg, ogeo, myslab, &Gs[cur][0], grp, l16);
    }
    wait_async0();
    __syncthreads();
  }

  // ---- combine + store (B,T,D) head-major ----
  float gt[8];
#pragma unroll
  for (int r = 0; r < 8; ++r) gt[r] = gate[b * TT + qbase + r + 8 * grp];
#pragma unroll
  for (int nt = 0; nt < 4; ++nt) {
#pragma unroll
    for (int r = 0; r < 8; ++r) {
      int row = qbase + r + 8 * grp;
      float o1 = ostd[nt][r] / ls[r];
      float o2 = ogeo[nt][r] / lg[r];
      float val = (1.f - gt[r]) * o1 + gt[r] * o2;
      comb[(size_t)(b * TT + row) * DDIM + h * DH + nt * 16 + l16] = val;
    }
  }
}

// ---------------------------------------------------------------------------
// Launch
// ---------------------------------------------------------------------------
extern "C" void kernel_launch(void* const* d_in, const int* in_sizes, int n_in,
                              void* d_out, int out_size, void* d_ws,
                              size_t ws_size, hipStream_t stream) {
  (void)in_sizes; (void)n_in; (void)out_size; (void)ws_size;
  const float* x = (const float*)d_in[0];
  const float* qkv_w = (const float*)d_in[1];
  const float* qkv_b = (const float*)d_in[2];
  const float* w1_write = (const float*)d_in[3];
  const float* w2_write = (const float*)d_in[4];
  const float* w1_read = (const float*)d_in[5];
  const float* w2_read = (const float*)d_in[6];
  const float* geo_w = (const float*)d_in[7];
  const float* geo_b = (const float*)d_in[8];
  const float* gate_w = (const float*)d_in[9];
  const float* gate_b = (const float*)d_in[10];
  const float* inc_scale = (const float*)d_in[11];
  const float* out_w = (const float*)d_in[12];
  const float* out_b = (const float*)d_in[13];
  float* out = (float*)d_out;

  // workspace layout (floats), total ~92.6 MB
  float* ws = (float*)d_ws;
  float* wq = ws;                      // (B,H,T,dh)   q   (k,v follow)
  float* wgeov = ws + 3 * SZ_HEAD;     // (B,H,T,dh)   geo_v
  float* wp1 = ws + 4 * SZ_HEAD;       // (B*T,64)     x @ w1_write
  float* ww2 = wp1 + (size_t)BT * 64;  // (B*T,64)     x @ w2_write
  float* wr1 = ww2 + (size_t)BT * 64;  // (B*T,64)     x @ w1_read
  float* wr2 = wr1 + (size_t)BT * 64;  // (B*T,64)     x @ w2_read
  float* wgpre = wr2 + (size_t)BT * 64;            // (B*T,16) gate pre-act
  float* wread8 = wgpre + (size_t)BT * 16;         // (B,H,T,8)
  float* wjw8 = wread8 + (size_t)BB * HH * TT * 8; // (B,H,T,8)
  float* wgate = wjw8 + (size_t)BB * HH * TT * 8;  // (B*T)
  float* wcomb = wgate + (size_t)BT;               // (B*T,D)

  const dim3 blk(256);

  // 1) qkv GEMM, scattered directly to (which,B,H,T,dh)
  gemm_f32_wmma<<<dim3(3072 / 64, BT / 128), blk, 0, stream>>>(
      x, qkv_w, qkv_b, wq, BT, DDIM, 3 * DDIM, 1);
  // 2-5) line projections (no bias)
  gemm_f32_wmma<<<dim3(1, BT / 128), blk, 0, stream>>>(x, w1_write, nullptr,
                                                       wp1, BT, DDIM, 64, 0);
  gemm_f32_wmma<<<dim3(1, BT / 128), blk, 0, stream>>>(x, w2_write, nullptr,
                                                       ww2, BT, DDIM, 64, 0);
  gemm_f32_wmma<<<dim3(1, BT / 128), blk, 0, stream>>>(x, w1_read, nullptr,
                                                       wr1, BT, DDIM, 64, 0);
  gemm_f32_wmma<<<dim3(1, BT / 128), blk, 0, stream>>>(x, w2_read, nullptr,
                                                       wr2, BT, DDIM, 64, 0);
  // 6) geo_v, scattered to (B,H,T,dh)
  gemm_f32_wmma<<<dim3(DDIM / 64, BT / 128), blk, 0, stream>>>(
      x, geo_w, geo_b, wgeov, BT, DDIM, DDIM, 2);
  // 7) gate pre-activation
  gemm_f32_wmma<<<dim3(1, BT / 128), blk, 0, stream>>>(x, gate_w, gate_b,
                                                       wgpre, BT, DDIM, 16, 0);
  // 8) Pluecker lines + J6 + gate
  prep_lines_gate<<<dim3(BT), dim3(32), 0, stream>>>(wp1, ww2, wr1, wr2, wgpre,
                                                     wjw8, wread8, wgate);
  // 9) dual-path causal flash attention (double-buffered async tile staging)
  attn_dual<<<dim3(BB * HH, TT / 128), blk, 0, stream>>>(
      wq, wq + SZ_HEAD, wq + 2 * SZ_HEAD, wgeov, wread8, wjw8, wgate, inc_scale,
      wcomb);
  // 10) output projection
  gemm_f32_wmma<<<dim3(DDIM / 64, BT / 128), blk, 0, stream>>>(
      wcomb, out_w, out_b, out, BT, DDIM, DDIM, 0);
}